// NWHead_12086037971171
// MI455X (gfx1250) — compile-verified
//
#include <hip/hip_runtime.h>
#include <hip/hip_bf16.h>
#include <math.h>

typedef __attribute__((ext_vector_type(2))) float v2f;
typedef __attribute__((ext_vector_type(4))) float v4f;
typedef __attribute__((ext_vector_type(8))) float v8f;

#define B_  64
#define N_  4096
#define D_  1024
#define C_  1000
#define NSPLIT 8
#define NCHUNK (N_ / NSPLIT)   // 512

// ---------------------------------------------------------------------------
// Kernel 1: scores[b][n] = -sqrt(max(||q||^2 - 2 q.s + ||s||^2, 0))
// One wave32 per 16-row tile of support rows; q.s via V_WMMA_F32_16X16X4_F32
// with the query chunk replicated across all 16 N-columns of B.
// All 8 waves of a block share one batch b: q[b] staged in LDS, ||q||^2
// computed once per block. Support rows streamed with NT hint (read-once).
// ---------------------------------------------------------------------------
__global__ __launch_bounds__(256)
void nw_scores_kernel(const float* __restrict__ q,     // (B,D)
                      const float* __restrict__ s,     // (B,N,D)
                      float* __restrict__ scores)      // (B,N)
{
    __shared__ float qsh[D_];          // staged query vector (4 KB)
    __shared__ float red[256];         // ||q||^2 reduction
    __shared__ float lds_dot[8][16];   // per-wave dot publication

    const int t    = threadIdx.x;
    const int lane = t & 31;
    const int wave = t >> 5;

    // 32 blocks per batch -> b is uniform across the block (scalar-friendly)
    const int b       = blockIdx.x >> 5;
    const int tileInB = ((blockIdx.x & 31) << 3) + wave;   // 0..255
    const int rowBase = tileInB << 4;
    const int hi = lane >> 4;                              // K-half select
    const int r  = lane & 15;                              // row within tile

    // ---- stage q[b] into LDS + block-wide ||q||^2 reduction ----
    {
        v4f qv = *(const v4f*)(q + (size_t)b * D_ + t * 4);
        *(v4f*)(qsh + t * 4) = qv;
        red[t] = qv.x * qv.x + qv.y * qv.y + qv.z * qv.z + qv.w * qv.w;
    }
    __syncthreads();
    for (int off = 128; off > 0; off >>= 1) {
        if (t < off) red[t] += red[t + off];
        __syncthreads();
    }
    const float q2 = red[0];

    const float* __restrict__ srow = s + ((size_t)b * N_ + rowBase + r) * D_;

    v8f   c   = {};                    // 16x16 f32 accumulator
    float s2a = 0.f;                   // partial ||s||^2 (this lane's K half)

    for (int k0 = 0; k0 < D_; k0 += 8) {
        // A: lanes 0-15 carry K = k0..k0+3, lanes 16-31 carry K = k0+4..k0+7
        v4f a  = __builtin_nontemporal_load((const v4f*)(srow + k0 + 4 * hi));
        v4f q0 = *(const v4f*)(qsh + k0);       // q[k0..k0+3]   (LDS broadcast)
        v4f q1 = *(const v4f*)(qsh + k0 + 4);   // q[k0+4..k0+7] (LDS broadcast)

        // wmma #1: K-slots map to global indices {k0, k0+1, k0+4, k0+5}
        // A layout (ISA 16x4 f32): VGPR0 = K0|K2, VGPR1 = K1|K3
        v2f A1 = { a.x, a.y };
        // B layout: VGPR0 = row K0 (lanes 0-15) | row K1 (lanes 16-31); VGPR1 = K2|K3
        v2f B1 = { hi ? q0.y : q0.x, hi ? q1.y : q1.x };
        c = __builtin_amdgcn_wmma_f32_16x16x4_f32(false, A1, false, B1,
                                                  (short)0, c, false, false);

        // wmma #2: K-slots map to {k0+2, k0+3, k0+6, k0+7}
        v2f A2 = { a.z, a.w };
        v2f B2 = { hi ? q0.w : q0.z, hi ? q1.w : q1.z };
        c = __builtin_amdgcn_wmma_f32_16x16x4_f32(false, A2, false, B2,
                                                  (short)0, c, false, false);

        s2a += a.x * a.x + a.y * a.y + a.z * a.z + a.w * a.w;
    }

    // ||s_r||^2 = lane-half r  +  lane-half r+16
    float s2 = s2a + __shfl_xor(s2a, 16, 32);

    // C layout: VGPR v = row v (lanes 0-15, N=lane) and row v+8 (lanes 16-31).
    // All N columns are identical, so lane 0 / lane 16 each publish 8 rows.
    if (lane == 0) {
#pragma unroll
        for (int v = 0; v < 8; ++v) lds_dot[wave][v] = c[v];
    }
    if (lane == 16) {
#pragma unroll
        for (int v = 0; v < 8; ++v) lds_dot[wave][8 + v] = c[v];
    }
    __syncthreads();

    if (lane < 16) {
        float dot = lds_dot[wave][lane];
        float d2  = fmaxf(q2 - 2.f * dot + s2, 0.f);
        scores[(size_t)b * N_ + rowBase + lane] = -sqrtf(d2);
    }
}

// ---------------------------------------------------------------------------
// Kernel 2: softmax over n (4096) per batch, in place. One block per batch.
// ---------------------------------------------------------------------------
__global__ __launch_bounds__(256)
void nw_softmax_kernel(float* __restrict__ scores)     // (B,N) -> probs
{
    __shared__ float red[256];
    const int b = blockIdx.x;
    const int t = threadIdx.x;
    float* p = scores + (size_t)b * N_;

    float vals[16];
    float m = -INFINITY;
#pragma unroll
    for (int i = 0; i < 16; ++i) {
        vals[i] = p[t + i * 256];
        m = fmaxf(m, vals[i]);
    }
    red[t] = m;
    __syncthreads();
    for (int off = 128; off > 0; off >>= 1) {
        if (t < off) red[t] = fmaxf(red[t], red[t + off]);
        __syncthreads();
    }
    m = red[0];
    __syncthreads();

    float sum = 0.f;
#pragma unroll
    for (int i = 0; i < 16; ++i) {
        vals[i] = __expf(vals[i] - m);
        sum += vals[i];
    }
    red[t] = sum;
    __syncthreads();
    for (int off = 128; off > 0; off >>= 1) {
        if (t < off) red[t] += red[t + off];
        __syncthreads();
    }
    float inv = 1.f / red[0];
#pragma unroll
    for (int i = 0; i < 16; ++i) p[t + i * 256] = vals[i] * inv;
}

// ---------------------------------------------------------------------------
// Kernel 3: partial[ns][b][c] = sum_{n in chunk ns} probs[b][n]*labels[b][n][c]
// probs chunk staged in LDS; 250 threads x float4 cover all 1000 classes with
// coalesced NT B128 label loads. No atomics -> deterministic.
// ---------------------------------------------------------------------------
__global__ __launch_bounds__(256)
void nw_wsum_kernel(const float* __restrict__ probs,   // (B,N)
                    const float* __restrict__ labels,  // (B,N,C)
                    float* __restrict__ partials)      // (NSPLIT,B,C)
{
    __shared__ float pl[NCHUNK];
    const int b  = blockIdx.x >> 3;                    // / NSPLIT
    const int ns = blockIdx.x & (NSPLIT - 1);
    const int t  = threadIdx.x;
    const int n0 = ns * NCHUNK;

    for (int i = t; i < NCHUNK; i += 256)
        pl[i] = probs[(size_t)b * N_ + n0 + i];
    __syncthreads();

    if (t < C_ / 4) {                                  // 250 threads
        const float* lab = labels + ((size_t)b * N_ + n0) * C_ + t * 4;
        v4f acc = {};
        for (int n = 0; n < NCHUNK; ++n) {
            float p = pl[n];
            v4f l = __builtin_nontemporal_load((const v4f*)(lab + (size_t)n * C_));
            acc.x += p * l.x;
            acc.y += p * l.y;
            acc.z += p * l.z;
            acc.w += p * l.w;
        }
        *(v4f*)(partials + (size_t)ns * B_ * C_ + (size_t)b * C_ + t * 4) = acc;
    }
}

// ---------------------------------------------------------------------------
// Kernel 4: out[b][c] = sum_ns partial[ns][b][c]
// ---------------------------------------------------------------------------
__global__ __launch_bounds__(256)
void nw_reduce_kernel(const float* __restrict__ partials, // (NSPLIT,B,C)
                      float* __restrict__ out)            // (B,C)
{
    int i = blockIdx.x * 256 + threadIdx.x;
    if (i < B_ * C_) {
        float a = 0.f;
#pragma unroll
        for (int k = 0; k < NSPLIT; ++k) a += partials[(size_t)k * B_ * C_ + i];
        out[i] = a;
    }
}

extern "C" void kernel_launch(void* const* d_in, const int* in_sizes, int n_in,
                              void* d_out, int out_size, void* d_ws, size_t ws_size,
                              hipStream_t stream) {
    const float* q   = (const float*)d_in[0];   // (64,1024)
    const float* s   = (const float*)d_in[1];   // (64,4096,1024)
    const float* lab = (const float*)d_in[2];   // (64,4096,1000)
    float* out = (float*)d_out;                 // (64,1000)

    float* probs    = (float*)d_ws;             // B_*N_ floats      (1 MB)
    float* partials = probs + (size_t)B_ * N_;  // NSPLIT*B_*C_ floats (~2 MB)

    // 16384 row-tiles, 8 waves/block -> 2048 blocks
    nw_scores_kernel<<<(B_ * (N_ / 16)) / 8, 256, 0, stream>>>(q, s, probs);
    nw_softmax_kernel<<<B_, 256, 0, stream>>>(probs);
    nw_wsum_kernel<<<B_ * NSPLIT, 256, 0, stream>>>(probs, lab, partials);
    nw_reduce_kernel<<<(B_ * C_ + 255) / 256, 256, 0, stream>>>(partials, out);
}